// LaneLoss_53163105190107
// MI455X (gfx1250) — compile-verified
//
#include <hip/hip_runtime.h>
#include <math.h>

// ---------------- problem constants (match reference) ----------------
#define BB 16
#define QQ 16
#define GG 6
#define PP 24
#define TT 7
#define RH 72
#define RW 128
#define NPX (RH*RW)          // 9216
#define MATCH_N 64
#define LOSS_N 96
#define THICK 0.03f
#define SHARP 80.0f

typedef __attribute__((ext_vector_type(2))) float v2f;
typedef __attribute__((ext_vector_type(8))) float v8f;

// ---------------- workspace layout (float offsets) ----------------
// all regions written before read; total ~3.37M floats (~13.5 MB), L2-resident
#define OFF_GCOMP 0                          // 96*48        = 4608
#define OFF_GR64  (OFF_GCOMP + 96*48)        // 96*128       = 12288
#define OFF_GR96  (OFF_GR64  + 96*128)       // 96*192       = 18432
#define OFF_PR64  (OFF_GR96  + 96*192)       // 256*128      = 32768
#define OFF_PR96  (OFF_PR64  + 256*128)      // 256*192      = 49152
#define OFF_PM    (OFF_PR96  + 256*192)      // 256*9216     = 2359296
#define OFF_GM    (OFF_PM    + 256*NPX)      // 96*9216      = 884736
#define OFF_COST  (OFF_GM    + 96*NPX)       // 1536
#define OFF_PAIR  (OFF_COST  + BB*QQ*GG)     // 96
#define OFF_PI    (OFF_PAIR  + BB*GG)        // 96 ints

// ---------------- helpers ----------------
__device__ __forceinline__ float sigmoidf(float x) { return 1.0f / (1.0f + __expf(-x)); }
__device__ __forceinline__ float bcel(float x, float t) {
    return fmaxf(x, 0.0f) - x * t + log1pf(__expf(-fabsf(x)));
}

__device__ __forceinline__ v8f wmma_k4(v2f a, v2f b) {
    v8f c = {0.f,0.f,0.f,0.f,0.f,0.f,0.f,0.f};
    // D(16x16,f32) = A(16x4,f32) * B(4x16,f32) + 0   -> v_wmma_f32_16x16x4_f32
    return __builtin_amdgcn_wmma_f32_16x16x4_f32(false, a, false, b, (short)0, c, false, false);
}

// one resampled point: inputs are 24-pt polyline + cumulative arclength
__device__ __forceinline__ void resample_one(const float* comp, const float* cum,
                                             float total, int num, int i,
                                             float& ox, float& oy) {
    if (total < 1e-8f) { ox = comp[0]; oy = comp[1]; return; }
    float t = (float)i * (total / (float)(num - 1));
    int idx = 0;
    #pragma unroll
    for (int j = 1; j < PP; j++) if (cum[j] <= t) idx = j;
    if (idx > PP - 2) idx = PP - 2;
    float alpha = (t - cum[idx]) / fmaxf(cum[idx + 1] - cum[idx], 1e-8f);
    ox = comp[2*idx]   + alpha * (comp[2*idx+2] - comp[2*idx]);
    oy = comp[2*idx+1] + alpha * (comp[2*idx+3] - comp[2*idx+1]);
}

// min distance from (x,y) to polyline (n points) held in LDS
__device__ __forceinline__ float p2pl_min(const float* poly, int n, float x, float y) {
    float best = 1e30f;
    for (int j = 0; j < n - 1; j++) {
        float ax = poly[2*j], ay = poly[2*j+1];
        float abx = poly[2*j+2] - ax, aby = poly[2*j+3] - ay;
        float t = ((x-ax)*abx + (y-ay)*aby) / fmaxf(abx*abx + aby*aby, 1e-8f);
        t = fminf(fmaxf(t, 0.0f), 1.0f);
        float dx = x - (ax + t*abx), dy = y - (ay + t*aby);
        best = fminf(best, dx*dx + dy*dy);
    }
    return sqrtf(best);
}

__device__ __forceinline__ void tangent_at(const float* r, int n, int i, float& tx, float& ty) {
    float fx, fy;
    if (i == 0)          { fx = r[2] - r[0];             fy = r[3] - r[1]; }
    else if (i == n - 1) { fx = r[2*i]   - r[2*i-2];     fy = r[2*i+1] - r[2*i-1]; }
    else                 { fx = r[2*i+2] - r[2*i-2];     fy = r[2*i+3] - r[2*i-1]; }
    float nrm = fmaxf(sqrtf(fx*fx + fy*fy), 1e-8f);
    tx = fx / nrm; ty = fy / nrm;
}

// ---------------- K1: GT compact + resample (96 blocks x 96 thr) ----------------
__global__ void k_gt_prep(const float* __restrict__ gpts, const float* __restrict__ gvis,
                          float* __restrict__ gcomp, float* __restrict__ gr64,
                          float* __restrict__ gr96) {
    __shared__ float comp[PP*2];
    __shared__ float cum[PP];
    __shared__ float s_total;
    int id = blockIdx.x;             // b*G+g
    int tid = threadIdx.x;
    if (tid == 0) {
        const float* pts = gpts + (size_t)id * PP * 2;
        const float* vis = gvis + (size_t)id * PP;
        float cx[PP], cy[PP];
        int cnt = 0;
        for (int j = 0; j < PP; j++)
            if (vis[j] > 0.5f) { cx[cnt] = pts[2*j]; cy[cnt] = pts[2*j+1]; cnt++; }
        if (cnt >= 2) {
            for (int j = cnt; j < PP; j++) { cx[j] = cx[cnt-1]; cy[j] = cy[cnt-1]; }
        } else {
            for (int j = 0; j < PP; j++) { cx[j] = pts[2*j]; cy[j] = pts[2*j+1]; }
        }
        float c = 0.f; cum[0] = 0.f;
        for (int j = 0; j < PP - 1; j++) {
            float dx = cx[j+1]-cx[j], dy = cy[j+1]-cy[j];
            c += sqrtf(dx*dx + dy*dy);
            cum[j+1] = c;
        }
        s_total = c;
        for (int j = 0; j < PP; j++) { comp[2*j] = cx[j]; comp[2*j+1] = cy[j]; }
    }
    __syncthreads();
    if (tid < PP*2) gcomp[(size_t)id * PP * 2 + tid] = comp[tid];
    float ox, oy;
    if (tid < MATCH_N) {
        resample_one(comp, cum, s_total, MATCH_N, tid, ox, oy);
        gr64[((size_t)id * MATCH_N + tid) * 2]     = ox;
        gr64[((size_t)id * MATCH_N + tid) * 2 + 1] = oy;
    }
    if (tid < LOSS_N) {
        resample_one(comp, cum, s_total, LOSS_N, tid, ox, oy);
        gr96[((size_t)id * LOSS_N + tid) * 2]     = ox;
        gr96[((size_t)id * LOSS_N + tid) * 2 + 1] = oy;
    }
}

// ---------------- K2: pred resample (256 blocks x 96 thr) ----------------
__global__ void k_pred_resample(const float* __restrict__ pred,
                                float* __restrict__ pr64, float* __restrict__ pr96) {
    __shared__ float comp[PP*2];
    __shared__ float cum[PP];
    __shared__ float s_total;
    int id = blockIdx.x;             // b*Q+q
    int tid = threadIdx.x;
    if (tid < PP*2) comp[tid] = pred[(size_t)id * PP * 2 + tid];
    __syncthreads();
    if (tid == 0) {
        float c = 0.f; cum[0] = 0.f;
        for (int j = 0; j < PP - 1; j++) {
            float dx = comp[2*j+2]-comp[2*j], dy = comp[2*j+3]-comp[2*j+1];
            c += sqrtf(dx*dx + dy*dy);
            cum[j+1] = c;
        }
        s_total = c;
    }
    __syncthreads();
    float ox, oy;
    if (tid < MATCH_N) {
        resample_one(comp, cum, s_total, MATCH_N, tid, ox, oy);
        pr64[((size_t)id * MATCH_N + tid) * 2]     = ox;
        pr64[((size_t)id * MATCH_N + tid) * 2 + 1] = oy;
    }
    if (tid < LOSS_N) {
        resample_one(comp, cum, s_total, LOSS_N, tid, ox, oy);
        pr96[((size_t)id * LOSS_N + tid) * 2]     = ox;
        pr96[((size_t)id * LOSS_N + tid) * 2 + 1] = oy;
    }
}

// ---------------- K3: soft masks via WMMA (352 blocks x 256 thr) ----------------
// dist^2(pixel,seg) = |p|^2 + (-2p.a + |a|^2) - 2*t*(ap.ab) + t^2*|ab|^2,
//   t = clamp((ap.ab)*inv|ab|^2, 0, 1)
// Both (ap.ab) and (-2p.a+|a|^2) are 16px x 16seg rank-3 matmuls:
//   A = [px py 1 0] (16x4), B1 = [abx; aby; -a.ab; 0], B2 = [-2ax; -2ay; |a|^2; 0]
// -> v_wmma_f32_16x16x4_f32. Segment-min is a 16-lane-half shfl_xor reduction.
__global__ void k_masks(const float* __restrict__ pred, const float* __restrict__ gcomp,
                        float* __restrict__ pm, float* __restrict__ gm) {
    __shared__ float s_abx[32], s_aby[32], s_nadab[32], s_n2ax[32], s_n2ay[32],
                     s_asq[32], s_abab[32], s_inv[32];
    int mid = blockIdx.x;
    const float* pts;
    float* out;
    if (mid < BB*QQ) { pts = pred  + (size_t)mid * PP * 2;            out = pm + (size_t)mid * NPX; }
    else             { int g = mid - BB*QQ;
                       pts = gcomp + (size_t)g   * PP * 2;            out = gm + (size_t)g   * NPX; }
    int tid  = threadIdx.x;
    if (tid < 32) {
        if (tid < PP - 1) {
            float ax = pts[2*tid],   ay = pts[2*tid+1];
            float bx = pts[2*tid+2], by = pts[2*tid+3];
            float abx = bx - ax, aby = by - ay;
            float abab = abx*abx + aby*aby;
            s_abx[tid] = abx; s_aby[tid] = aby;
            s_nadab[tid] = -(ax*abx + ay*aby);
            s_n2ax[tid] = -2.f*ax; s_n2ay[tid] = -2.f*ay;
            s_asq[tid]  = ax*ax + ay*ay;
            s_abab[tid] = abab;
            s_inv[tid]  = 1.f / fmaxf(abab, 1e-8f);
        } else {  // padded segments: never win the min (d2 ~ 1e30)
            s_abx[tid] = 0.f; s_aby[tid] = 0.f; s_nadab[tid] = 0.f;
            s_n2ax[tid] = 0.f; s_n2ay[tid] = 0.f; s_asq[tid] = 1e30f;
            s_abab[tid] = 0.f; s_inv[tid] = 0.f;
        }
    }
    __syncthreads();

    int lane = tid & 31, wave = tid >> 5;
    int half = lane >> 4;          // 0: lanes 0-15 (K0/K1), 1: lanes 16-31 (K2/K3)
    int nseg = lane & 15;          // N index of this lane in B/D

    // hoist per-lane B operands for both 16-seg tiles
    v2f b1[2], b2[2];
    float abab_t[2], inv_t[2];
    #pragma unroll
    for (int s = 0; s < 2; s++) {
        int sg = s*16 + nseg;
        v2f t1, t2;
        if (half == 0) { t1.x = s_abx[sg];   t1.y = s_aby[sg];
                         t2.x = s_n2ax[sg];  t2.y = s_n2ay[sg]; }
        else           { t1.x = s_nadab[sg]; t1.y = 0.f;
                         t2.x = s_asq[sg];   t2.y = 0.f; }
        b1[s] = t1; b2[s] = t2;
        abab_t[s] = s_abab[sg]; inv_t[s] = s_inv[sg];
    }

    const float inv127 = 1.f/127.f, inv71 = 1.f/71.f;
    // 576 pixel tiles of 16; 8 waves -> 72 uniform iterations each (EXEC full at WMMA)
    for (int tile = wave; tile < NPX/16; tile += 8) {
        int basepx = tile * 16;
        int row = basepx >> 7;       // RW = 128
        int x0  = basepx & 127;
        float py = (float)row * inv71;

        v2f a;
        if (half == 0) { a.x = (float)(x0 + nseg) * inv127; a.y = py; }  // [px py]
        else           { a.x = 1.f; a.y = 0.f; }                          // [1 0]

        float psq[8];
        #pragma unroll
        for (int c = 0; c < 8; c++) {
            float px = (float)(x0 + c + half*8) * inv127;   // pixel M = c + 8*half
            psq[c] = px*px + py*py;
        }

        v8f u0  = wmma_k4(a, b1[0]);   // ap.ab, segs 0-15
        v8f s20 = wmma_k4(a, b2[0]);   // -2p.a+|a|^2, segs 0-15
        v8f u1  = wmma_k4(a, b1[1]);   // segs 16-31 (23..31 padded)
        v8f s21 = wmma_k4(a, b2[1]);

        float dmin[8];
        #pragma unroll
        for (int c = 0; c < 8; c++) {
            float uu = u0[c];
            float t  = fminf(fmaxf(uu * inv_t[0], 0.f), 1.f);
            float d2a = psq[c] + s20[c] - 2.f*t*uu + t*t*abab_t[0];
            uu = u1[c];
            t  = fminf(fmaxf(uu * inv_t[1], 0.f), 1.f);
            float d2b = psq[c] + s21[c] - 2.f*t*uu + t*t*abab_t[1];
            dmin[c] = fminf(d2a, d2b);
        }
        // min over segments = min across the 16 lanes of this half
        #pragma unroll
        for (int off = 1; off < 16; off <<= 1) {
            #pragma unroll
            for (int c = 0; c < 8; c++)
                dmin[c] = fminf(dmin[c], __shfl_xor(dmin[c], off, 32));
        }
        if (nseg == 0) {   // lane 0 -> pixels M=0..7, lane 16 -> M=8..15
            int po = basepx + half * 8;
            #pragma unroll
            for (int c = 0; c < 8; c++) {
                float d = sqrtf(fmaxf(dmin[c], 0.f));
                out[po + c] = sigmoidf((THICK - d) * SHARP);
            }
        }
    }
}

// ---------------- K4: cost matrix (1536 blocks x 256 thr) ----------------
__global__ void k_cost(const float* __restrict__ pr64, const float* __restrict__ gr64,
                       const float* __restrict__ pm, const float* __restrict__ gm,
                       const float* __restrict__ exist, float* __restrict__ cost) {
    __shared__ float spr[MATCH_N*2], sgr[MATCH_N*2];
    __shared__ float a_pg, a_gp, a_tan, a_curv, a_inter, a_umax;
    int blk = blockIdx.x;
    int b = blk / (QQ*GG), rem = blk % (QQ*GG);
    int q = rem / GG, g = rem % GG;
    int tid = threadIdx.x;

    if (tid < MATCH_N*2) {
        spr[tid] = pr64[((size_t)(b*QQ+q) * MATCH_N) * 2 + tid];
        sgr[tid] = gr64[((size_t)(b*GG+g) * MATCH_N) * 2 + tid];
    }
    if (tid == 0) { a_pg=0.f; a_gp=0.f; a_tan=0.f; a_curv=0.f; a_inter=0.f; a_umax=0.f; }
    __syncthreads();

    float l = 0.f;
    if (tid < MATCH_N) {                         // pr -> gr
        l = p2pl_min(sgr, MATCH_N, spr[2*tid], spr[2*tid+1]);
        atomicAdd(&a_pg, l);
        float txp, typ, txg, tyg;
        tangent_at(spr, MATCH_N, tid, txp, typ);
        tangent_at(sgr, MATCH_N, tid, txg, tyg);
        atomicAdd(&a_tan, fabsf(txp*txg + typ*tyg));
    } else if (tid < 2*MATCH_N) {                // gr -> pr
        int i = tid - MATCH_N;
        l = p2pl_min(spr, MATCH_N, sgr[2*i], sgr[2*i+1]);
        atomicAdd(&a_gp, l);
    }
    if (tid < MATCH_N - 2) {                     // curvature (Huber)
        int i = tid + 2;
        float cv = 0.f;
        #pragma unroll
        for (int c = 0; c < 2; c++) {
            float dp = spr[2*i+c] - 2.f*spr[2*(i-1)+c] + spr[2*(i-2)+c];
            float dg = sgr[2*i+c] - 2.f*sgr[2*(i-1)+c] + sgr[2*(i-2)+c];
            float d = fabsf(dp - dg);
            cv += (d < 1.f) ? 0.5f*d*d : d - 0.5f;
        }
        atomicAdd(&a_curv, cv);
    }
    // mask IoU (min/max form) — L2-resident stream; prefetch one stride ahead
    float li = 0.f, lu = 0.f;
    const float* pmb = pm + (size_t)(b*QQ+q) * NPX;
    const float* gmb = gm + (size_t)(b*GG+g) * NPX;
    for (int px = tid; px < NPX; px += 256) {
        if (px + 256 < NPX) {
            __builtin_prefetch(pmb + px + 256, 0, 0);   // global_prefetch_b8
            __builtin_prefetch(gmb + px + 256, 0, 0);
        }
        float pv = pmb[px], gv = gmb[px];
        li += fminf(pv, gv);
        lu += fmaxf(pv, gv);
    }
    atomicAdd(&a_inter, li);
    atomicAdd(&a_umax, lu);
    __syncthreads();

    if (tid == 0) {
        float sym  = 0.5f * (a_pg / MATCH_N + a_gp / MATCH_N);
        float tanv = 1.f - a_tan / MATCH_N;
        float curv = a_curv / (float)((MATCH_N - 2) * 2);
        float iou  = a_inter / fmaxf(a_umax, 1e-6f);
        float ex   = sigmoidf(exist[b*QQ + q]);
        cost[(size_t)(b*QQ+q) * GG + g] =
            sym + 0.35f*tanv + 0.2f*curv + 0.75f*(1.f - iou) - 0.25f*ex;
    }
}

// ---------------- K5: Hungarian (JV), 16 blocks x 1 thr ----------------
__global__ void k_hungarian(const float* __restrict__ cost, int* __restrict__ pi) {
    int b = blockIdx.x;
    const int n = GG, m = QQ;
    double c[GG][QQ];
    for (int i = 0; i < n; i++)
        for (int j = 0; j < m; j++)
            c[i][j] = (double)cost[(size_t)(b*QQ + j) * GG + i];   // transposed
    double u[GG+1] = {0}, v[QQ+1] = {0};
    int p[QQ+1] = {0}, way[QQ+1] = {0};
    for (int i = 1; i <= n; i++) {
        p[0] = i; int j0 = 0;
        double minv[QQ+1]; bool used[QQ+1];
        for (int j = 0; j <= m; j++) { minv[j] = 1e18; used[j] = false; }
        do {
            used[j0] = true;
            int i0 = p[j0], j1 = 0;
            double delta = 1e18;
            for (int j = 1; j <= m; j++) if (!used[j]) {
                double cur = c[i0-1][j-1] - u[i0] - v[j];
                if (cur < minv[j]) { minv[j] = cur; way[j] = j0; }
                if (minv[j] < delta) { delta = minv[j]; j1 = j; }
            }
            for (int j = 0; j <= m; j++) {
                if (used[j]) { u[p[j]] += delta; v[j] -= delta; }
                else         { minv[j] -= delta; }
            }
            j0 = j1;
        } while (p[j0] != 0);
        while (j0) { int j1 = way[j0]; p[j0] = p[j1]; j0 = j1; }
    }
    for (int j = 1; j <= m; j++)
        if (p[j] != 0) pi[b*GG + (p[j]-1)] = j - 1;
}

// ---------------- K6: per-pair loss (96 blocks x 256 thr) ----------------
__global__ void k_pair_loss(const float* __restrict__ pr96, const float* __restrict__ gr96,
                            const float* __restrict__ pm, const float* __restrict__ gm,
                            const float* __restrict__ typel, const float* __restrict__ visl,
                            const int* __restrict__ gtype, const float* __restrict__ gvis,
                            const int* __restrict__ pi, float* __restrict__ pair_out) {
    __shared__ float spr[LOSS_N*2], sgr[LOSS_N*2];
    __shared__ float a_pg, a_gp, a_tan, a_curv, a_inter, a_sp, a_sg, a_vis;
    int blk = blockIdx.x;
    int b = blk / GG, g = blk % GG;
    int q = pi[b*GG + g];
    int tid = threadIdx.x;

    if (tid < LOSS_N*2) {
        spr[tid] = pr96[((size_t)(b*QQ+q) * LOSS_N) * 2 + tid];
        sgr[tid] = gr96[((size_t)(b*GG+g) * LOSS_N) * 2 + tid];
    }
    if (tid == 0) { a_pg=0.f; a_gp=0.f; a_tan=0.f; a_curv=0.f;
                    a_inter=0.f; a_sp=0.f; a_sg=0.f; a_vis=0.f; }
    __syncthreads();

    if (tid < LOSS_N) {
        float l = p2pl_min(sgr, LOSS_N, spr[2*tid], spr[2*tid+1]);
        atomicAdd(&a_pg, l);
        float txp, typ2, txg, tyg;
        tangent_at(spr, LOSS_N, tid, txp, typ2);
        tangent_at(sgr, LOSS_N, tid, txg, tyg);
        atomicAdd(&a_tan, fabsf(txp*txg + typ2*tyg));
    } else if (tid < 2*LOSS_N) {
        int i = tid - LOSS_N;
        float l = p2pl_min(spr, LOSS_N, sgr[2*i], sgr[2*i+1]);
        atomicAdd(&a_gp, l);
    }
    if (tid < LOSS_N - 2) {
        int i = tid + 2;
        float cv = 0.f;
        #pragma unroll
        for (int c = 0; c < 2; c++) {
            float dp = spr[2*i+c] - 2.f*spr[2*(i-1)+c] + spr[2*(i-2)+c];
            float dg = sgr[2*i+c] - 2.f*sgr[2*(i-1)+c] + sgr[2*(i-2)+c];
            float d = fabsf(dp - dg);
            cv += (d < 1.f) ? 0.5f*d*d : d - 0.5f;
        }
        atomicAdd(&a_curv, cv);
    }
    // Dice/IoU mask sums (product form) — L2-resident stream; prefetch ahead
    float li = 0.f, lp = 0.f, lg = 0.f;
    const float* pmb = pm + (size_t)(b*QQ+q) * NPX;
    const float* gmb = gm + (size_t)(b*GG+g) * NPX;
    for (int px = tid; px < NPX; px += 256) {
        if (px + 256 < NPX) {
            __builtin_prefetch(pmb + px + 256, 0, 0);   // global_prefetch_b8
            __builtin_prefetch(gmb + px + 256, 0, 0);
        }
        float pv = pmb[px], gv = gmb[px];
        li += pv * gv; lp += pv; lg += gv;
    }
    atomicAdd(&a_inter, li); atomicAdd(&a_sp, lp); atomicAdd(&a_sg, lg);
    // visibility BCE
    if (tid < PP) {
        float x = visl[(size_t)(b*QQ+q) * PP + tid];
        float t = gvis[(size_t)(b*GG+g) * PP + tid];
        atomicAdd(&a_vis, bcel(x, t));
    }
    __syncthreads();

    if (tid == 0) {
        float sym  = 0.5f * (a_pg / LOSS_N + a_gp / LOSS_N);
        float tanv = 1.f - a_tan / LOSS_N;
        float curv = a_curv / (float)((LOSS_N - 2) * 2);
        float uni  = fmaxf(a_sp + a_sg - a_inter, 1e-6f);
        float overlap = 0.5f * ((1.f - a_inter / uni) +
                                (1.f - (2.f*a_inter + 1e-6f) / (a_sp + a_sg + 1e-6f)));
        // CE on type logits
        const float* tl = typel + (size_t)(b*QQ+q) * TT;
        float mx = tl[0];
        for (int k = 1; k < TT; k++) mx = fmaxf(mx, tl[k]);
        float se = 0.f;
        for (int k = 0; k < TT; k++) se += __expf(tl[k] - mx);
        float lse = mx + logf(se);
        float ce = lse - tl[gtype[b*GG + g]];
        float visl_mean = a_vis / (float)PP;
        const float invn = 1.f / (float)(BB*GG);
        pair_out[blk] = (5.0f*sym + 1.0f*tanv + 0.5f*curv +
                         2.0f*overlap + 1.0f*ce + visl_mean) * invn;
    }
}

// ---------------- K7: final reduction (1 block x 256 thr) ----------------
__global__ void k_final(const float* __restrict__ exist, const int* __restrict__ pi,
                        const float* __restrict__ pair_out, float* __restrict__ out) {
    __shared__ float s_bce, s_pair;
    int tid = threadIdx.x;
    if (tid == 0) { s_bce = 0.f; s_pair = 0.f; }
    __syncthreads();
    // existence BCE over all (b,q)
    int b = tid / QQ, q = tid % QQ;
    float tgt = 0.f;
    #pragma unroll
    for (int g = 0; g < GG; g++) if (pi[b*GG + g] == q) tgt = 1.f;
    atomicAdd(&s_bce, bcel(exist[b*QQ + q], tgt));
    if (tid < BB*GG) atomicAdd(&s_pair, pair_out[tid]);
    __syncthreads();
    if (tid == 0)
        out[0] = 2.0f * (s_bce / (float)(BB*QQ)) + s_pair;
}

// ---------------- host ----------------
extern "C" void kernel_launch(void* const* d_in, const int* in_sizes, int n_in,
                              void* d_out, int out_size, void* d_ws, size_t ws_size,
                              hipStream_t stream) {
    const float* exist = (const float*)d_in[0];   // (B,Q,1)
    const float* pred  = (const float*)d_in[1];   // (B,Q,P,2)
    const float* typel = (const float*)d_in[2];   // (B,Q,T)
    const float* visl  = (const float*)d_in[3];   // (B,Q,P)
    const float* gpts  = (const float*)d_in[5];   // (B,G,P,2)
    const float* gvis  = (const float*)d_in[6];   // (B,G,P)
    const int*   gtype = (const int*)d_in[7];     // (B,G)

    float* ws    = (float*)d_ws;
    float* gcomp = ws + OFF_GCOMP;
    float* gr64  = ws + OFF_GR64;
    float* gr96  = ws + OFF_GR96;
    float* pr64  = ws + OFF_PR64;
    float* pr96  = ws + OFF_PR96;
    float* pm    = ws + OFF_PM;
    float* gm    = ws + OFF_GM;
    float* cost  = ws + OFF_COST;
    float* pair  = ws + OFF_PAIR;
    int*   pi    = (int*)(ws + OFF_PI);

    k_gt_prep<<<BB*GG, 96, 0, stream>>>(gpts, gvis, gcomp, gr64, gr96);
    k_pred_resample<<<BB*QQ, 96, 0, stream>>>(pred, pr64, pr96);
    k_masks<<<BB*QQ + BB*GG, 256, 0, stream>>>(pred, gcomp, pm, gm);
    k_cost<<<BB*QQ*GG, 256, 0, stream>>>(pr64, gr64, pm, gm, exist, cost);
    k_hungarian<<<BB, 1, 0, stream>>>(cost, pi);
    k_pair_loss<<<BB*GG, 256, 0, stream>>>(pr96, gr96, pm, gm, typel, visl,
                                           gtype, gvis, pi, pair);
    k_final<<<1, 256, 0, stream>>>(exist, pi, pair, (float*)d_out);
}